// GNN_15642270892473
// MI455X (gfx1250) — compile-verified
//
#include <hip/hip_runtime.h>

// ---------------------------------------------------------------------------
// CDNA5 (gfx1250) implementation of the TAGConv GNN reference.
// Dense 64-wide GEMMs use V_WMMA_F32_16X16X4_F32 (fp32 matrix pipe, wave32).
// Sparse propagation uses fp32 global atomics (h tiles are L2-resident).
// ---------------------------------------------------------------------------

typedef __attribute__((ext_vector_type(2))) float v2f;
typedef __attribute__((ext_vector_type(8))) float v8f;

#define LDS_STRIDE 66   // 16x64 tile padded: avoids LDS bank conflicts between lane halves

__device__ __forceinline__ v8f wmma_f32_16x16x4(v2f a, v2f b, v8f c) {
    // 8-arg form: (neg_a, A, neg_b, B, c_mod, C, reuse_a, reuse_b)
    return __builtin_amdgcn_wmma_f32_16x16x4_f32(false, a, false, b, (short)0, c,
                                                 false, false);
}

__device__ __forceinline__ float leaky(float v) {
    return v >= 0.0f ? v : 0.01f * v;
}

// ---------------------------------------------------------------------------
// Fused MLP encoder: h = leaky(leaky(x@W1 + b1)@W2 + b2)
// One wave per 16-node tile; intermediate tile staged through LDS to convert
// WMMA C-layout -> A-layout. 8 waves / block.
// ---------------------------------------------------------------------------
__global__ void enc_kernel(const float* __restrict__ x,
                           const float* __restrict__ W1, const float* __restrict__ b1,
                           const float* __restrict__ W2, const float* __restrict__ b2,
                           float* __restrict__ h, int n)
{
    __shared__ float lds[8][16 * LDS_STRIDE];
    const int lane = threadIdx.x & 31;
    const int wave = threadIdx.x >> 5;
    const int tile = blockIdx.x * 8 + wave;
    if (tile * 16 >= n) return;          // wave-uniform guard (EXEC stays all-ones)

    const int half = lane >> 4;          // K-pair select for A/B frags
    const int q    = lane & 15;          // row (A) / col (B,C) within tile

    const float* X = x + (size_t)tile * 16 * 8;
    float* t = lds[wave];

    v8f acc[4];
    #pragma unroll
    for (int nt = 0; nt < 4; ++nt)
        #pragma unroll
        for (int r = 0; r < 8; ++r) acc[nt][r] = 0.0f;

    // ---- stage 1: [16x8] @ [8x64], K=8 -> 2 WMMA steps ----
    #pragma unroll
    for (int kk = 0; kk < 2; ++kk) {
        const int k0 = kk * 4 + half * 2;
        v2f a = *(const v2f*)(X + q * 8 + k0);         // A: (m=q, K=k0..k0+1)
        #pragma unroll
        for (int nt = 0; nt < 4; ++nt) {
            const int colb = nt * 16 + q;
            v2f b;
            b.x = W1[(k0 + 0) * 64 + colb];            // B: (K, N=colb)
            b.y = W1[(k0 + 1) * 64 + colb];
            acc[nt] = wmma_f32_16x16x4(a, b, acc[nt]);
        }
    }

    // bias + leaky, spill C-layout -> LDS (row-major 16x64 tile, stride 66)
    #pragma unroll
    for (int nt = 0; nt < 4; ++nt) {
        const int colb = nt * 16 + q;
        const float bv = b1[colb];
        #pragma unroll
        for (int r = 0; r < 8; ++r) {
            const int m = r + 8 * half;                // C: (M=r+8*half, N=colb)
            t[m * LDS_STRIDE + colb] = leaky(acc[nt][r] + bv);
        }
    }

    // ---- stage 2: [16x64] @ [64x64], K=64 -> 16 WMMA steps ----
    #pragma unroll
    for (int nt = 0; nt < 4; ++nt)
        #pragma unroll
        for (int r = 0; r < 8; ++r) acc[nt][r] = 0.0f;

    #pragma unroll 4
    for (int kk = 0; kk < 16; ++kk) {
        const int k0 = kk * 4 + half * 2;
        v2f a = *(const v2f*)(t + q * LDS_STRIDE + k0);
        #pragma unroll
        for (int nt = 0; nt < 4; ++nt) {
            const int colb = nt * 16 + q;
            v2f b;
            b.x = W2[(k0 + 0) * 64 + colb];
            b.y = W2[(k0 + 1) * 64 + colb];
            acc[nt] = wmma_f32_16x16x4(a, b, acc[nt]);
        }
    }

    #pragma unroll
    for (int nt = 0; nt < 4; ++nt) {
        const int colb = nt * 16 + q;
        const float bv = b2[colb];
        #pragma unroll
        for (int r = 0; r < 8; ++r) {
            const int m = tile * 16 + r + 8 * half;
            h[(size_t)m * 64 + colb] = leaky(acc[nt][r] + bv);
        }
    }
}

// ---------------------------------------------------------------------------
// 64x64 GEMM over node tiles: Out(+)= Hin @ W  (+bias) (leaky optional)
// accumulate=1 preloads existing Out into the WMMA accumulator.
// ---------------------------------------------------------------------------
__global__ void gemm64_kernel(const float* __restrict__ Hin, const float* __restrict__ W,
                              float* __restrict__ Out, const float* __restrict__ bias,
                              int accumulate, int do_leaky, int n)
{
    __shared__ float lds[8][16 * LDS_STRIDE];
    const int lane = threadIdx.x & 31;
    const int wave = threadIdx.x >> 5;
    const int tile = blockIdx.x * 8 + wave;
    if (tile * 16 >= n) return;

    const int half = lane >> 4;
    const int q    = lane & 15;

    // coalesced stage of the 16x64 A tile into LDS (256 float4s / 32 lanes)
    const float* A = Hin + (size_t)tile * 16 * 64;
    float* t = lds[wave];
    #pragma unroll
    for (int i = 0; i < 8; ++i) {
        const int idx = lane + i * 32;                 // float4 index in tile
        const int r   = idx >> 4;
        const int c4  = (idx & 15) << 2;
        const float4 v = *(const float4*)(A + r * 64 + c4);
        float* d = t + r * LDS_STRIDE + c4;
        d[0] = v.x; d[1] = v.y; d[2] = v.z; d[3] = v.w;
    }

    v8f acc[4];
    #pragma unroll
    for (int nt = 0; nt < 4; ++nt) {
        const int colb = nt * 16 + q;
        #pragma unroll
        for (int r = 0; r < 8; ++r) {
            acc[nt][r] = accumulate
                ? Out[(size_t)(tile * 16 + r + 8 * half) * 64 + colb]
                : 0.0f;
        }
    }

    #pragma unroll 4
    for (int kk = 0; kk < 16; ++kk) {
        const int k0 = kk * 4 + half * 2;
        v2f a = *(const v2f*)(t + q * LDS_STRIDE + k0);
        #pragma unroll
        for (int nt = 0; nt < 4; ++nt) {
            const int colb = nt * 16 + q;
            v2f b;
            b.x = W[(k0 + 0) * 64 + colb];
            b.y = W[(k0 + 1) * 64 + colb];
            acc[nt] = wmma_f32_16x16x4(a, b, acc[nt]);
        }
    }

    #pragma unroll
    for (int nt = 0; nt < 4; ++nt) {
        const int colb = nt * 16 + q;
        const float bv = bias ? bias[colb] : 0.0f;
        #pragma unroll
        for (int r = 0; r < 8; ++r) {
            float v = acc[nt][r] + bv;
            if (do_leaky) v = leaky(v);
            Out[(size_t)(tile * 16 + r + 8 * half) * 64 + colb] = v;
        }
    }
}

// ---------------------------------------------------------------------------
// gcn_norm pieces
// ---------------------------------------------------------------------------
__global__ void deg_kernel(const int* __restrict__ col, const float* __restrict__ ew,
                           float* __restrict__ deg, int e)
{
    int i = blockIdx.x * blockDim.x + threadIdx.x;
    if (i >= e) return;
    atomicAdd(&deg[col[i]], ew[i]);
}

__global__ void dinv_kernel(const float* __restrict__ deg, float* __restrict__ dinv, int n)
{
    int i = blockIdx.x * blockDim.x + threadIdx.x;
    if (i >= n) return;
    const float d = deg[i];
    dinv[i] = d > 0.0f ? rsqrtf(d) : 0.0f;
}

__global__ void norm_kernel(const int* __restrict__ row, const int* __restrict__ col,
                            const float* __restrict__ ew, const float* __restrict__ dinv,
                            float* __restrict__ w, int e)
{
    int i = blockIdx.x * blockDim.x + threadIdx.x;
    if (i >= e) return;
    w[i] = dinv[row[i]] * ew[i] * dinv[col[i]];
}

// ---------------------------------------------------------------------------
// One propagation hop: hout[col[e]] += w[e] * hin[row[e]]   (64 features)
// 8 threads per edge, each owning 8 features (2x float4 gathers from L2).
// ---------------------------------------------------------------------------
__global__ void prop_kernel(const int* __restrict__ row, const int* __restrict__ col,
                            const float* __restrict__ w, const float* __restrict__ hin,
                            float* __restrict__ hout, int e)
{
    const int gid  = blockIdx.x * blockDim.x + threadIdx.x;
    const int edge = gid >> 3;
    if (edge >= e) return;
    const int f0 = (gid & 7) * 8;

    const int   r  = row[edge];
    const int   c  = col[edge];
    const float we = w[edge];

    const float4 v0 = *(const float4*)(hin + (size_t)r * 64 + f0);
    const float4 v1 = *(const float4*)(hin + (size_t)r * 64 + f0 + 4);
    float* dst = hout + (size_t)c * 64 + f0;
    atomicAdd(dst + 0, we * v0.x);
    atomicAdd(dst + 1, we * v0.y);
    atomicAdd(dst + 2, we * v0.z);
    atomicAdd(dst + 3, we * v0.w);
    atomicAdd(dst + 4, we * v1.x);
    atomicAdd(dst + 5, we * v1.y);
    atomicAdd(dst + 6, we * v1.z);
    atomicAdd(dst + 7, we * v1.w);
}

// ---------------------------------------------------------------------------
// Decoder second layer: [N,64] @ [64,2] + b  (tiny; plain VALU)
// ---------------------------------------------------------------------------
__global__ void dec2_kernel(const float* __restrict__ t, const float* __restrict__ W,
                            const float* __restrict__ b, float* __restrict__ out, int n)
{
    int i = blockIdx.x * blockDim.x + threadIdx.x;
    if (i >= n) return;
    const float* r = t + (size_t)i * 64;
    float a0 = b[0], a1 = b[1];
    #pragma unroll 8
    for (int k = 0; k < 64; ++k) {
        const float v = r[k];
        a0 += v * W[k * 2 + 0];
        a1 += v * W[k * 2 + 1];
    }
    out[i * 2 + 0] = a0;
    out[i * 2 + 1] = a1;
}

// ---------------------------------------------------------------------------
extern "C" void kernel_launch(void* const* d_in, const int* in_sizes, int n_in,
                              void* d_out, int out_size, void* d_ws, size_t ws_size,
                              hipStream_t stream)
{
    const float* x  = (const float*)d_in[0];
    const int*   ei = (const int*)d_in[1];
    const float* ew = (const float*)d_in[2];
    const int n = in_sizes[0] / 8;       // F_IN = 8
    const int e = in_sizes[1] / 2;       // edge_index is [2, E]
    const int* row = ei;
    const int* col = ei + e;

    auto F = [&](int i) { return (const float*)d_in[i]; };

    const float *We1, *be1, *We2, *be2, *Wd1, *bd1, *Wd2, *bd2;
    const float *Wg[2][4];
    const float *bg[2];

    if (in_sizes[3] == 8 * 64) {
        // insertion-order flatten: enc, gnn (lins then bias), dec
        We1 = F(3);  be1 = F(4);  We2 = F(5);  be2 = F(6);
        Wg[0][0] = F(7);  Wg[0][1] = F(8);  Wg[0][2] = F(9);  Wg[0][3] = F(10); bg[0] = F(11);
        Wg[1][0] = F(12); Wg[1][1] = F(13); Wg[1][2] = F(14); Wg[1][3] = F(15); bg[1] = F(16);
        Wd1 = F(17); bd1 = F(18); Wd2 = F(19); bd2 = F(20);
    } else {
        // sorted pytree flatten: dec, enc, gnn (bias before lins)
        Wd1 = F(3);  bd1 = F(4);  Wd2 = F(5);  bd2 = F(6);
        We1 = F(7);  be1 = F(8);  We2 = F(9);  be2 = F(10);
        bg[0] = F(11); Wg[0][0] = F(12); Wg[0][1] = F(13); Wg[0][2] = F(14); Wg[0][3] = F(15);
        bg[1] = F(16); Wg[1][0] = F(17); Wg[1][1] = F(18); Wg[1][2] = F(19); Wg[1][3] = F(20);
    }

    // workspace carve-up (all 256B aligned since n*64*4 is)
    const size_t NB = (size_t)n * 64 * sizeof(float);
    char* ws = (char*)d_ws;
    float* h    = (float*)(ws);
    float* tA   = (float*)(ws + NB);
    float* tB   = (float*)(ws + 2 * NB);
    float* outb = (float*)(ws + 3 * NB);
    float* deg  = (float*)(ws + 4 * NB);
    float* dinv = deg + n;
    float* wn   = dinv + n;

    const int tileGrid = (n + 127) / 128;                 // 8 waves x 16 nodes
    const int eGrid    = (e + 255) / 256;
    const int nGrid    = (n + 255) / 256;
    const int propGrid = (int)((8LL * e + 255) / 256);

    // ---- encoder + gcn_norm ----
    hipMemsetAsync(deg, 0, (size_t)n * sizeof(float), stream);
    enc_kernel<<<tileGrid, 256, 0, stream>>>(x, We1, be1, We2, be2, h, n);
    deg_kernel<<<eGrid, 256, 0, stream>>>(col, ew, deg, e);
    dinv_kernel<<<nGrid, 256, 0, stream>>>(deg, dinv, n);
    norm_kernel<<<eGrid, 256, 0, stream>>>(row, col, ew, dinv, wn, e);

    // ---- two TAGConv layers ----
    float* hin = h;
    float* ob  = outb;
    for (int l = 0; l < 2; ++l) {
        // out = h @ W0
        gemm64_kernel<<<tileGrid, 256, 0, stream>>>(hin, Wg[l][0], ob, nullptr, 0, 0, n);
        const float* cur = hin;
        for (int k = 1; k <= 3; ++k) {
            float* dst = (k & 1) ? tA : tB;
            hipMemsetAsync(dst, 0, NB, stream);
            prop_kernel<<<propGrid, 256, 0, stream>>>(row, col, wn, cur, dst, e);
            const int last = (k == 3);
            gemm64_kernel<<<tileGrid, 256, 0, stream>>>(
                dst, Wg[l][k], ob,
                last ? bg[l] : nullptr,       // bias once per layer
                1,                            // accumulate
                (last && l == 0) ? 1 : 0,     // leaky between layers only
                n);
            cur = dst;
        }
        float* tmp = hin; hin = ob; ob = tmp;
    }

    // ---- decoder ----
    gemm64_kernel<<<tileGrid, 256, 0, stream>>>(hin, Wd1, tA, bd1, 0, 1, n);
    dec2_kernel<<<nGrid, 256, 0, stream>>>(tA, Wd2, bd2, (float*)d_out, n);
}